// InstanceModule_884763263722
// MI455X (gfx1250) — compile-verified
//
#include <hip/hip_runtime.h>
#include <math.h>

typedef __attribute__((ext_vector_type(2))) float v2f;
typedef __attribute__((ext_vector_type(8))) float v8f;

// ---------------------------------------------------------------------------
// zero-fill scratch (graph-capture safe; re-run every call for determinism)
// ---------------------------------------------------------------------------
__global__ void k_zero4(float4* __restrict__ p, long n4) {
    long i = (long)blockIdx.x * blockDim.x + threadIdx.x;
    if (i < n4) p[i] = make_float4(0.f, 0.f, 0.f, 0.f);
}
__global__ void k_zero(float* __restrict__ p, long n) {
    long i = (long)blockIdx.x * blockDim.x + threadIdx.x;
    if (i < n) p[i] = 0.0f;
}

// ---------------------------------------------------------------------------
// Block-1 node transform (6 -> 32), three outputs in one pass:
//   base = x @ (W_self1 + W_res1) + b1   (self + residual-proj folded)
//   ta   = x @ W_tp1
//   tb   = x @ W_int1
// K=6 is too small for WMMA; plain FMA, one thread per (node, channel).
// ---------------------------------------------------------------------------
__global__ void k_node1(const float* __restrict__ x,
                        const float* __restrict__ Wself, const float* __restrict__ b1,
                        const float* __restrict__ Wtp,   const float* __restrict__ Wint,
                        const float* __restrict__ Wres,
                        float* __restrict__ base, float* __restrict__ ta,
                        float* __restrict__ tb, int N) {
    int i = blockIdx.x * blockDim.x + threadIdx.x;
    if (i >= N * 32) return;
    int n = i >> 5, c = i & 31;
    const float* xr = x + (size_t)n * 6;
    float s_base = b1[c], s_tp = 0.f, s_int = 0.f;
#pragma unroll
    for (int k = 0; k < 6; ++k) {
        float xv = xr[k];
        s_base += xv * (Wself[k * 32 + c] + Wres[k * 32 + c]);
        s_tp   += xv * Wtp[k * 32 + c];
        s_int  += xv * Wint[k * 32 + c];
    }
    base[i] = s_base;
    ta[i]   = s_tp;
    tb[i]   = s_int;
}

// ---------------------------------------------------------------------------
// Edge aggregation: m[dst] += t[src]  (32 ch, 8 threads/edge, float4 gather,
// hardware global_atomic_add_f32 scatter into L2-resident accumulators).
// Optional degree count (for the 'intersects' mean reducer).
// ---------------------------------------------------------------------------
__global__ void k_edge_agg(const int* __restrict__ src, const int* __restrict__ dst,
                           const float* __restrict__ t, float* __restrict__ m,
                           float* __restrict__ cnt, int E) {
    int tid = blockIdx.x * blockDim.x + threadIdx.x;
    int e = tid >> 3, q = tid & 7;
    if (e >= E) return;
    int s = src[e], d = dst[e];
    const float4 v = *(const float4*)(t + (size_t)s * 32 + q * 4);
    float* mp = m + (size_t)d * 32 + q * 4;
    unsafeAtomicAdd(mp + 0, v.x);
    unsafeAtomicAdd(mp + 1, v.y);
    unsafeAtomicAdd(mp + 2, v.z);
    unsafeAtomicAdd(mp + 3, v.w);
    if (cnt != nullptr && q == 0) unsafeAtomicAdd(cnt + d, 1.0f);
}

// ---------------------------------------------------------------------------
// Combine passes (elementwise, float4-vectorized): mean-normalize, add, ReLU.
// One thread per (node, 4 channels): i in [0, N*8).
// ---------------------------------------------------------------------------
__global__ void k_combine1(const float4* __restrict__ base, const float4* __restrict__ mtp,
                           const float4* __restrict__ mint, const float* __restrict__ cnt,
                           float4* __restrict__ h, int N) {
    int i = blockIdx.x * blockDim.x + threadIdx.x;
    if (i >= N * 8) return;
    int n = i >> 3;
    float inv = 1.0f / fmaxf(cnt[n], 1.0f);
    float4 b = base[i], t = mtp[i], m = mint[i], r;
    r.x = fmaxf(b.x + t.x + m.x * inv, 0.f);
    r.y = fmaxf(b.y + t.y + m.y * inv, 0.f);
    r.z = fmaxf(b.z + t.z + m.z * inv, 0.f);
    r.w = fmaxf(b.w + t.w + m.w * inv, 0.f);
    h[i] = r;
}

// Block 2 combine: identity residual (+h1), bias b2, in-place on h.
__global__ void k_combine2(const float4* __restrict__ tself, const float* __restrict__ b2,
                           const float4* __restrict__ mtp, const float4* __restrict__ mint,
                           const float* __restrict__ cnt, float4* __restrict__ h, int N) {
    int i = blockIdx.x * blockDim.x + threadIdx.x;
    if (i >= N * 8) return;
    int n = i >> 3;
    float inv = 1.0f / fmaxf(cnt[n], 1.0f);
    float4 bb = ((const float4*)b2)[i & 7];
    float4 s = tself[i], t = mtp[i], m = mint[i], hv = h[i], r;
    r.x = fmaxf(s.x + bb.x + hv.x + t.x + m.x * inv, 0.f);
    r.y = fmaxf(s.y + bb.y + hv.y + t.y + m.y * inv, 0.f);
    r.z = fmaxf(s.z + bb.z + hv.z + t.z + m.z * inv, 0.f);
    r.w = fmaxf(s.w + bb.w + hv.w + t.w + m.w * inv, 0.f);
    h[i] = r;
}

// ---------------------------------------------------------------------------
// WMMA fp32 GEMM: out[16 nodes, 32] = A[16 nodes, 32] @ W[32, 32]
// One wave32 per 16-node group.  V_WMMA_F32_16X16X4_F32, K=32 -> 8 steps,
// N=32 -> 2 column tiles.  Layouts per CDNA5 ISA 7.12.2:
//   A 16x4 f32:  lanes 0-15 rows M, VGPR0/1 = K0/K1; lanes 16-31 = K2/K3
//   B 4x16 f32:  mirrored (VGPR0/1 = K0,K2 / K1,K3 across lane halves)
//   C/D 16x16:   VGPR r -> row r (lanes 0-15) / row r+8 (lanes 16-31)
// Store guard is wave-uniform (full-tile fast path; N=100000 is 16-divisible
// so the guarded path is never taken in this problem).
// ---------------------------------------------------------------------------
__device__ __forceinline__ void load_a_frag(const float* __restrict__ h, int node0,
                                            int lane, int N, v2f a[8]) {
    int row = node0 + (lane & 15);
    if (row >= N) row = N - 1;                 // safe duplicate; never stored
    int kb = (lane >> 4) * 2;                  // lanes 16-31 hold K+2,K+3
    const float* p = h + (size_t)row * 32 + kb;
#pragma unroll
    for (int j = 0; j < 8; ++j) {
        v2f t;
        t.x = p[4 * j + 0];
        t.y = p[4 * j + 1];
        a[j] = t;
    }
}

__device__ __forceinline__ void wmma_gemm_store(const v2f a[8], const float* __restrict__ W,
                                                float* __restrict__ out, int node0,
                                                int lane, int N) {
    int cl = lane & 15;
    int hi = lane >> 4;
    const bool full = (node0 + 16 <= N);       // wave-uniform: one scalar branch
#pragma unroll
    for (int t = 0; t < 2; ++t) {              // two 16-wide column tiles
        int col = t * 16 + cl;
        v8f c = {};
#pragma unroll
        for (int j = 0; j < 8; ++j) {          // K = 32 in steps of 4
            int kb = 4 * j + hi * 2;
            v2f b;
            b.x = W[kb * 32 + col];
            b.y = W[(kb + 1) * 32 + col];
            c = __builtin_amdgcn_wmma_f32_16x16x4_f32(
                    false, a[j], false, b, (short)0, c, false, false);
        }
        int rbase = node0 + hi * 8;
        float* p = out + (size_t)rbase * 32 + col;
        if (full) {
#pragma unroll
            for (int r = 0; r < 8; ++r) p[(size_t)r * 32] = c[r];
        } else {
#pragma unroll
            for (int r = 0; r < 8; ++r)
                if (rbase + r < N) p[(size_t)r * 32] = c[r];
        }
    }
}

// Block-2 node transform: one A-fragment load, three weight matrices.
__global__ void k_gemm32x3(const float* __restrict__ h,
                           const float* __restrict__ W0, const float* __restrict__ W1,
                           const float* __restrict__ W2,
                           float* __restrict__ o0, float* __restrict__ o1,
                           float* __restrict__ o2, int N) {
    int lane = threadIdx.x;
    int node0 = blockIdx.x * 16;
    v2f a[8];
    load_a_frag(h, node0, lane, N, a);
    wmma_gemm_store(a, W0, o0, node0, lane, N);
    wmma_gemm_store(a, W1, o1, node0, lane, N);
    wmma_gemm_store(a, W2, o2, node0, lane, N);
}

// Decoder first layer: single 32x32 GEMM.
__global__ void k_gemm32x1(const float* __restrict__ h, const float* __restrict__ W,
                           float* __restrict__ o, int N) {
    int lane = threadIdx.x;
    int node0 = blockIdx.x * 16;
    v2f a[8];
    load_a_frag(h, node0, lane, N, a);
    wmma_gemm_store(a, W, o, node0, lane, N);
}

// ---------------------------------------------------------------------------
// Decoder tail: out = sigmoid( relu(td1 + bd1) @ Wd2 + bd2 )
// ---------------------------------------------------------------------------
__global__ void k_dec(const float* __restrict__ td1, const float* __restrict__ bd1,
                      const float* __restrict__ Wd2, const float* __restrict__ bd2,
                      float* __restrict__ out, int N) {
    int n = blockIdx.x * blockDim.x + threadIdx.x;
    if (n >= N) return;
    const float4* r4 = (const float4*)(td1 + (size_t)n * 32);
    float s = bd2[0];
#pragma unroll
    for (int q = 0; q < 8; ++q) {
        float4 v = r4[q];
        const float* bb = bd1 + q * 4;
        const float* ww = Wd2 + q * 4;
        s += fmaxf(v.x + bb[0], 0.f) * ww[0];
        s += fmaxf(v.y + bb[1], 0.f) * ww[1];
        s += fmaxf(v.z + bb[2], 0.f) * ww[2];
        s += fmaxf(v.w + bb[3], 0.f) * ww[3];
    }
    out[n] = 1.0f / (1.0f + __expf(-s));
}

// ---------------------------------------------------------------------------
extern "C" void kernel_launch(void* const* d_in, const int* in_sizes, int n_in,
                              void* d_out, int out_size, void* d_ws, size_t ws_size,
                              hipStream_t stream) {
    const float* x     = (const float*)d_in[0];
    const int*   etp   = (const int*)d_in[1];
    const int*   eint  = (const int*)d_in[2];
    const float* Ws1   = (const float*)d_in[3];
    const float* b1    = (const float*)d_in[4];
    const float* Wtp1  = (const float*)d_in[5];
    const float* Wint1 = (const float*)d_in[6];
    const float* Wres1 = (const float*)d_in[7];
    const float* Ws2   = (const float*)d_in[8];
    const float* b2    = (const float*)d_in[9];
    const float* Wtp2  = (const float*)d_in[10];
    const float* Wint2 = (const float*)d_in[11];
    const float* Wd1   = (const float*)d_in[12];
    const float* bd1   = (const float*)d_in[13];
    const float* Wd2   = (const float*)d_in[14];
    const float* bd2   = (const float*)d_in[15];

    const int N  = in_sizes[0] / 6;     // 100000
    const int E1 = in_sizes[1] / 2;     // 1600000
    const int E2 = in_sizes[2] / 2;

    const long NC = (long)N * 32;
    float* ws   = (float*)d_ws;
    float* base = ws;                   // base1 -> tself2 -> td1
    float* ta   = ws + 1 * NC;          // t_tp (blk1 then blk2)
    float* tb   = ws + 2 * NC;          // t_int
    float* mtp  = ws + 3 * NC;          // sum-agg accumulator
    float* mint = ws + 4 * NC;          // mean-agg accumulator
    float* h    = ws + 5 * NC;          // h1, then h2 in-place
    float* cnt  = ws + 6 * NC;          // N floats (degree, shared by blocks)
    float* out  = (float*)d_out;

    const int TB = 256;
    auto cdiv = [](long a, long b) { return (int)((a + b - 1) / b); };
    const int ngrp = (N + 15) / 16;     // 6250 waves for WMMA GEMMs

    // ---- block 1 ----
    k_zero4<<<cdiv(NC / 2, TB), TB, 0, stream>>>((float4*)mtp, NC / 2); // mtp+mint
    k_zero<<<cdiv(N, TB), TB, 0, stream>>>(cnt, (long)N);
    k_node1<<<cdiv(NC, TB), TB, 0, stream>>>(x, Ws1, b1, Wtp1, Wint1, Wres1,
                                             base, ta, tb, N);
    k_edge_agg<<<cdiv((long)E1 * 8, TB), TB, 0, stream>>>(etp, etp + E1, ta, mtp,
                                                          nullptr, E1);
    k_edge_agg<<<cdiv((long)E2 * 8, TB), TB, 0, stream>>>(eint, eint + E2, tb, mint,
                                                          cnt, E2);
    k_combine1<<<cdiv((long)N * 8, TB), TB, 0, stream>>>((const float4*)base,
                                                         (const float4*)mtp,
                                                         (const float4*)mint, cnt,
                                                         (float4*)h, N);

    // ---- block 2 ----
    k_zero4<<<cdiv(NC / 2, TB), TB, 0, stream>>>((float4*)mtp, NC / 2);
    k_gemm32x3<<<ngrp, 32, 0, stream>>>(h, Ws2, Wtp2, Wint2, base, ta, tb, N);
    k_edge_agg<<<cdiv((long)E1 * 8, TB), TB, 0, stream>>>(etp, etp + E1, ta, mtp,
                                                          nullptr, E1);
    k_edge_agg<<<cdiv((long)E2 * 8, TB), TB, 0, stream>>>(eint, eint + E2, tb, mint,
                                                          nullptr, E2);   // cnt reused
    k_combine2<<<cdiv((long)N * 8, TB), TB, 0, stream>>>((const float4*)base, b2,
                                                         (const float4*)mtp,
                                                         (const float4*)mint, cnt,
                                                         (float4*)h, N);

    // ---- decoder ----
    k_gemm32x1<<<ngrp, 32, 0, stream>>>(h, Wd1, base, N);
    k_dec<<<cdiv(N, TB), TB, 0, stream>>>(base, bd1, Wd2, bd2, out, N);
}